// VoxelEncoder_24524263260768
// MI455X (gfx1250) — compile-verified
//
#include <hip/hip_runtime.h>
#include <hip/hip_bf16.h>

// ---------------- constants ----------------
#define DD   500
#define HH   500
#define WW   40
#define HW_  (HH * WW)          // 20000
#define GRID_CELLS 10000000     // DD*HH*WW
#define MAXV 40000
#define MAXP 32
#define NROWS (MAXV * MAXP)     // 1,280,000
#define SCAN_BLK 4096
#define NSCAN ((GRID_CELLS + SCAN_BLK - 1) / SCAN_BLK)  // 2442
#define BN_EPS 1e-5f

typedef float v2f __attribute__((ext_vector_type(2)));
typedef float v8f __attribute__((ext_vector_type(8)));

__device__ __forceinline__ v8f wmma4(v2f a, v2f b, v8f c) {
    // V_WMMA_F32_16X16X4_F32 : D = A(16x4) * B(4x16) + C(16x16)
    return __builtin_amdgcn_wmma_f32_16x16x4_f32(
        false, a, false, b, (short)0, c, false, false);
}

__device__ __forceinline__ void wait_dscnt0() {
    asm volatile("s_wait_dscnt 0x0" ::: "memory");
}

// ---------------- voxelization ----------------
__device__ __forceinline__ bool bin_point(float4 p, int& key) {
    // matches ((xyz - RANGE_MIN)/VOXEL).astype(int32): trunc toward zero
    int ix = (int)((p.x + 50.0f) / 0.2f);
    int iy = (int)((p.y + 50.0f) / 0.2f);
    int iz = (int)((p.z + 3.0f)  / 0.2f);
    if (ix < 0 || ix >= DD || iy < 0 || iy >= HH || iz < 0 || iz >= WW) return false;
    key = ix * HW_ + iy * WW + iz;
    return true;
}

__global__ void mark_kernel(const float4* __restrict__ pts, int n, int* __restrict__ grid) {
    int i = blockIdx.x * blockDim.x + threadIdx.x;
    if (i >= n) return;
    int key;
    if (bin_point(pts[i], key)) grid[key] = 1;
}

__global__ void blockcount_kernel(const int* __restrict__ grid, int* __restrict__ bsums) {
    __shared__ int sc[256];
    int t = threadIdx.x;
    int base = blockIdx.x * SCAN_BLK + t * 16;
    int c = 0;
#pragma unroll
    for (int i = 0; i < 16; ++i) {
        int idx = base + i;
        if (idx < GRID_CELLS) c += grid[idx];
    }
    sc[t] = c;
    __syncthreads();
    for (int off = 128; off > 0; off >>= 1) {
        if (t < off) sc[t] += sc[t + off];
        __syncthreads();
    }
    if (t == 0) bsums[blockIdx.x] = sc[0];
}

__global__ void scan_blocks_kernel(int* __restrict__ bsums, int nblocks) {
    // single thread serial exclusive scan over ~2442 entries
    int run = 0;
    for (int b = 0; b < nblocks; ++b) {
        int v = bsums[b];
        bsums[b] = run;
        run += v;
    }
}

__global__ void assign_kernel(int* __restrict__ grid, const int* __restrict__ bsums,
                              float* __restrict__ coordsOut) {
    __shared__ int sc[256];
    int t = threadIdx.x;
    int base = blockIdx.x * SCAN_BLK + t * 16;
    int flags[16];
    int cnt = 0;
#pragma unroll
    for (int i = 0; i < 16; ++i) {
        int idx = base + i;
        flags[i] = (idx < GRID_CELLS) ? grid[idx] : 0;
        cnt += flags[i];
    }
    sc[t] = cnt;
    __syncthreads();
    // Hillis-Steele inclusive scan over 256 per-thread counts
    for (int off = 1; off < 256; off <<= 1) {
        int add = (t >= off) ? sc[t - off] : 0;
        __syncthreads();
        sc[t] += add;
        __syncthreads();
    }
    int rank = bsums[blockIdx.x] + sc[t] - cnt;   // exclusive offset
    for (int i = 0; i < 16; ++i) {
        int idx = base + i;
        if (idx >= GRID_CELLS) break;
        if (flags[i]) {
            int vid = (rank < MAXV) ? rank : -1;
            grid[idx] = vid;
            if (vid >= 0) {
                int z = idx / HW_;
                int y = (idx % HW_) / WW;
                int x = idx % WW;
                coordsOut[vid * 4 + 0] = 0.0f;
                coordsOut[vid * 4 + 1] = (float)z;
                coordsOut[vid * 4 + 2] = (float)y;
                coordsOut[vid * 4 + 3] = (float)x;
            }
            ++rank;
        } else {
            grid[idx] = -1;
        }
    }
}

__global__ void scatter_kernel(const float4* __restrict__ pts, int n,
                               const int* __restrict__ grid, int* __restrict__ counts,
                               float4* __restrict__ vf) {
    int i = blockIdx.x * blockDim.x + threadIdx.x;
    if (i >= n) return;
    float4 p = pts[i];
    int key;
    if (!bin_point(p, key)) return;
    int vid = grid[key];
    if (vid < 0) return;
    int slot = atomicAdd(&counts[vid], 1);
    if (slot < MAXP) vf[vid * MAXP + slot] = p;
}

// ---------------- MLP via WMMA f32 16x16x4 ----------------
// stats region (floats): [0:64) sum1 [64:128) sumsq1 [128:192) sum2 [192:256) sumsq2
//                        [256:320) sc1 [320:384) sh1 [384:448) sc2 [448:512) sh2

__global__ __launch_bounds__(128) void stats1_kernel(const float* __restrict__ vf,
                                                     const float* __restrict__ W1,
                                                     const float* __restrict__ b1,
                                                     float* __restrict__ acc) {
    int lane = threadIdx.x & 31;
    int gw   = (blockIdx.x * blockDim.x + threadIdx.x) >> 5;
    int rowbase = gw * 16;
    int m = lane & 15, half = lane >> 4, kb = half * 2;
    const float* xr = vf + (size_t)(rowbase + m) * 4;
    v2f a; a.x = xr[kb]; a.y = xr[kb + 1];
#pragma unroll
    for (int t = 0; t < 4; ++t) {
        int n = t * 16 + m;
        v2f b; b.x = W1[kb * 64 + n]; b.y = W1[(kb + 1) * 64 + n];
        v8f c;
#pragma unroll
        for (int r = 0; r < 8; ++r) c[r] = b1[n];
        c = wmma4(a, b, c);
        float s = 0.f, ss = 0.f;
#pragma unroll
        for (int r = 0; r < 8; ++r) { s += c[r]; ss += c[r] * c[r]; }
        s  += __shfl_xor(s, 16, 32);
        ss += __shfl_xor(ss, 16, 32);
        if (half == 0) {
            atomicAdd(&acc[n], s);
            atomicAdd(&acc[64 + n], ss);
        }
    }
}

__global__ void finalize_bn_kernel(const float* __restrict__ sum, const float* __restrict__ sumsq,
                                   const float* __restrict__ g, const float* __restrict__ be,
                                   float* __restrict__ sc, float* __restrict__ sh) {
    int c = threadIdx.x;
    if (c < 64) {
        const float invN = 1.0f / (float)NROWS;
        float mu  = sum[c] * invN;
        float var = sumsq[c] * invN - mu * mu;
        float rs  = rsqrtf(var + BN_EPS);
        float s   = rs * g[c];
        sc[c] = s;
        sh[c] = be[c] - mu * s;
    }
}

__global__ __launch_bounds__(128) void stats2_kernel(const float* __restrict__ vf,
                                                     const float* __restrict__ W1,
                                                     const float* __restrict__ b1,
                                                     const float* __restrict__ W2,
                                                     const float* __restrict__ b2,
                                                     float* __restrict__ acc) {
    __shared__ float lds[4 * 16 * 64];
    int lane = threadIdx.x & 31;
    int wid  = threadIdx.x >> 5;
    int gw   = blockIdx.x * 4 + wid;
    int rowbase = gw * 16;
    int m = lane & 15, half = lane >> 4, kb = half * 2;
    float* L = lds + wid * 1024;
    const float* sc1 = acc + 256;
    const float* sh1 = acc + 320;

    const float* xr = vf + (size_t)(rowbase + m) * 4;
    v2f a; a.x = xr[kb]; a.y = xr[kb + 1];
#pragma unroll
    for (int t = 0; t < 4; ++t) {
        int n = t * 16 + m;
        v2f b; b.x = W1[kb * 64 + n]; b.y = W1[(kb + 1) * 64 + n];
        v8f c;
#pragma unroll
        for (int r = 0; r < 8; ++r) c[r] = b1[n];
        c = wmma4(a, b, c);
        float s = sc1[n], h = sh1[n];
#pragma unroll
        for (int r = 0; r < 8; ++r) {
            float v = c[r] * s + h;
            L[(r + half * 8) * 64 + n] = v > 0.f ? v : 0.f;
        }
    }
    wait_dscnt0();

    v8f c2[4];
#pragma unroll
    for (int t = 0; t < 4; ++t) {
        int n = t * 16 + m;
#pragma unroll
        for (int r = 0; r < 8; ++r) c2[t][r] = b2[n];
    }
    for (int k0 = 0; k0 < 64; k0 += 4) {
        v2f aa; aa.x = L[m * 64 + k0 + kb]; aa.y = L[m * 64 + k0 + kb + 1];
#pragma unroll
        for (int t = 0; t < 4; ++t) {
            int n = t * 16 + m;
            v2f bb; bb.x = W2[(k0 + kb) * 64 + n]; bb.y = W2[(k0 + kb + 1) * 64 + n];
            c2[t] = wmma4(aa, bb, c2[t]);
        }
    }
#pragma unroll
    for (int t = 0; t < 4; ++t) {
        int n = t * 16 + m;
        float s = 0.f, ss = 0.f;
#pragma unroll
        for (int r = 0; r < 8; ++r) { s += c2[t][r]; ss += c2[t][r] * c2[t][r]; }
        s  += __shfl_xor(s, 16, 32);
        ss += __shfl_xor(ss, 16, 32);
        if (half == 0) {
            atomicAdd(&acc[128 + n], s);
            atomicAdd(&acc[192 + n], ss);
        }
    }
}

__global__ __launch_bounds__(128) void final_kernel(const float* __restrict__ vf,
                                                    const float* __restrict__ W1,
                                                    const float* __restrict__ b1,
                                                    const float* __restrict__ W2,
                                                    const float* __restrict__ b2,
                                                    const float* __restrict__ W3,
                                                    const float* __restrict__ b3,
                                                    const float* __restrict__ prm,
                                                    const int* __restrict__ counts,
                                                    float* __restrict__ out) {
    __shared__ float lds[4 * 16 * 64];
    __shared__ float mbuf[2 * 128];
    int lane = threadIdx.x & 31;
    int wid  = threadIdx.x >> 5;
    int gw   = blockIdx.x * 4 + wid;
    int rowbase = gw * 16;
    int m = lane & 15, half = lane >> 4, kb = half * 2;
    float* L = lds + wid * 1024;
    const float* sc1 = prm + 256; const float* sh1 = prm + 320;
    const float* sc2 = prm + 384; const float* sh2 = prm + 448;

    // layer1 -> a1 in LDS
    const float* xr = vf + (size_t)(rowbase + m) * 4;
    v2f a; a.x = xr[kb]; a.y = xr[kb + 1];
#pragma unroll
    for (int t = 0; t < 4; ++t) {
        int n = t * 16 + m;
        v2f b; b.x = W1[kb * 64 + n]; b.y = W1[(kb + 1) * 64 + n];
        v8f c;
#pragma unroll
        for (int r = 0; r < 8; ++r) c[r] = b1[n];
        c = wmma4(a, b, c);
        float s = sc1[n], h = sh1[n];
#pragma unroll
        for (int r = 0; r < 8; ++r) {
            float v = c[r] * s + h;
            L[(r + half * 8) * 64 + n] = v > 0.f ? v : 0.f;
        }
    }
    wait_dscnt0();

    // layer2 -> a2 in LDS (overwrite a1 after all reads)
    v8f c2[4];
#pragma unroll
    for (int t = 0; t < 4; ++t) {
        int n = t * 16 + m;
#pragma unroll
        for (int r = 0; r < 8; ++r) c2[t][r] = b2[n];
    }
    for (int k0 = 0; k0 < 64; k0 += 4) {
        v2f aa; aa.x = L[m * 64 + k0 + kb]; aa.y = L[m * 64 + k0 + kb + 1];
#pragma unroll
        for (int t = 0; t < 4; ++t) {
            int n = t * 16 + m;
            v2f bb; bb.x = W2[(k0 + kb) * 64 + n]; bb.y = W2[(k0 + kb + 1) * 64 + n];
            c2[t] = wmma4(aa, bb, c2[t]);
        }
    }
    wait_dscnt0();
#pragma unroll
    for (int t = 0; t < 4; ++t) {
        int n = t * 16 + m;
        float s = sc2[n], h = sh2[n];
#pragma unroll
        for (int r = 0; r < 8; ++r) {
            float v = c2[t][r] * s + h;
            L[(r + half * 8) * 64 + n] = v > 0.f ? v : 0.f;
        }
    }
    wait_dscnt0();

    // layer3: 8 N-tiles of 16 (128 channels)
    v8f c3[8];
#pragma unroll
    for (int t = 0; t < 8; ++t) {
        int n = t * 16 + m;
#pragma unroll
        for (int r = 0; r < 8; ++r) c3[t][r] = b3[n];
    }
    for (int k0 = 0; k0 < 64; k0 += 4) {
        v2f aa; aa.x = L[m * 64 + k0 + kb]; aa.y = L[m * 64 + k0 + kb + 1];
#pragma unroll
        for (int t = 0; t < 8; ++t) {
            int n = t * 16 + m;
            v2f bb; bb.x = W3[(k0 + kb) * 128 + n]; bb.y = W3[(k0 + kb + 1) * 128 + n];
            c3[t] = wmma4(aa, bb, c3[t]);
        }
    }

    // masked max over 16 rows of this tile
    int v  = rowbase >> 5;          // voxel id
    int tb = rowbase & 31;          // 0 or 16: slot tile base
    int cnt = counts[v]; if (cnt > MAXP) cnt = MAXP;
    const float NEGINF = -__builtin_huge_valf();
    float mx[8];
#pragma unroll
    for (int t = 0; t < 8; ++t) {
        float q = NEGINF;
#pragma unroll
        for (int r = 0; r < 8; ++r) {
            int slot = tb + r + half * 8;
            if (slot < cnt) q = fmaxf(q, c3[t][r]);
        }
        q = fmaxf(q, __shfl_xor(q, 16, 32));
        mx[t] = q;
    }
    int vb = wid >> 1;  // voxel within block (0/1)
    if (tb == 0 && half == 0) {
#pragma unroll
        for (int t = 0; t < 8; ++t) mbuf[vb * 128 + t * 16 + m] = mx[t];
    }
    __syncthreads();
    if (tb == 16 && half == 0) {
#pragma unroll
        for (int t = 0; t < 8; ++t) {
            int n = t * 16 + m;
            out[(size_t)v * 128 + n] = fmaxf(mx[t], mbuf[vb * 128 + n]);
        }
    }
}

// ---------------- launcher ----------------
extern "C" void kernel_launch(void* const* d_in, const int* in_sizes, int n_in,
                              void* d_out, int out_size, void* d_ws, size_t ws_size,
                              hipStream_t stream) {
    const float* pts = (const float*)d_in[0];
    const float* W1  = (const float*)d_in[1];
    const float* b1  = (const float*)d_in[2];
    const float* g1  = (const float*)d_in[3];
    const float* be1 = (const float*)d_in[4];
    const float* W2  = (const float*)d_in[5];
    const float* b2  = (const float*)d_in[6];
    const float* g2  = (const float*)d_in[7];
    const float* be2 = (const float*)d_in[8];
    const float* W3  = (const float*)d_in[9];
    const float* b3  = (const float*)d_in[10];
    int Npts = in_sizes[0] / 4;

    char* ws = (char*)d_ws;
    int*   grid   = (int*)  (ws + 0);            // 40,000,000 B
    int*   bsums  = (int*)  (ws + 40000000);     //     16,384 B
    int*   counts = (int*)  (ws + 40016384);     //    163,840 B
    float* stats  = (float*)(ws + 40180224);     //      2,048 B
    float* vf     = (float*)(ws + 40182272);     // 20,480,000 B
    const size_t WS_USED = 60662272;

    hipMemsetAsync(d_ws, 0, WS_USED, stream);
    hipMemsetAsync(d_out, 0, (size_t)out_size * sizeof(float), stream);

    float* voxel_out = (float*)d_out;                 // 40000*128
    float* coordsOut = (float*)d_out + MAXV * 128;    // 40000*4 (as floats)

    int pblocks = (Npts + 255) / 256;
    mark_kernel<<<pblocks, 256, 0, stream>>>((const float4*)pts, Npts, grid);
    blockcount_kernel<<<NSCAN, 256, 0, stream>>>(grid, bsums);
    scan_blocks_kernel<<<1, 1, 0, stream>>>(bsums, NSCAN);
    assign_kernel<<<NSCAN, 256, 0, stream>>>(grid, bsums, coordsOut);
    scatter_kernel<<<pblocks, 256, 0, stream>>>((const float4*)pts, Npts, grid, counts, (float4*)vf);

    const int nwaves  = NROWS / 16;      // 80,000
    const int nblocks = nwaves / 4;      // 20,000 blocks of 128 threads
    stats1_kernel<<<nblocks, 128, 0, stream>>>(vf, W1, b1, stats);
    finalize_bn_kernel<<<1, 64, 0, stream>>>(stats + 0,   stats + 64,  g1, be1, stats + 256, stats + 320);
    stats2_kernel<<<nblocks, 128, 0, stream>>>(vf, W1, b1, W2, b2, stats);
    finalize_bn_kernel<<<1, 64, 0, stream>>>(stats + 128, stats + 192, g2, be2, stats + 384, stats + 448);
    final_kernel<<<nblocks, 128, 0, stream>>>(vf, W1, b1, W2, b2, W3, b3, stats, counts, voxel_out);
}